// GroupedVectorAttention_2508260901453
// MI455X (gfx1250) — compile-verified
//
#include <hip/hip_runtime.h>
#include <math.h>

// ---------------------------------------------------------------------------
// GroupedVectorAttention (PointTransformer-v2 style) for MI455X / gfx1250.
// N=100000 points, K=16 neighbors, C=64 channels, G=8 groups. f32 throughout
// (reference is f32; we use V_WMMA_F32_16X16X4_F32 so the matrix pipe is used
// without precision loss).
//
// Pass structure (all BatchNorms are training-mode => global reductions):
//  k1: Xq/Xk/Vv = feat@W + b via f32 WMMA; per-channel sum/sumsq partials.
//  k2: pos 1st/2nd moments (9 scalars) -> closed-form BN for pos@wp1+bp1.
//  k3: finalize BN(q), BN(k), BN(pos-branch); fold pos BN into wp1/bp1.
//  k4: per point (1 wave / point): peb via WMMA (16x64 @ 64x64), y = (k-q+peb)@ww1+bw1
//      stored to ws; per-group sum/sumsq partials for BN(y).
//  k5: finalize BN(y).
//  k6: per point: peb via WMMA again, w = relu(bn(y))@ww2+bw2, grouped softmax
//      over K (mask applied AFTER softmax, like the reference), weighted sum.
//
// Determinism: no atomics anywhere. All reductions are shuffle + fixed-order
// LDS partials + fixed-order global partial sweep. Graph-capture safe: only
// kernel launches.
// ---------------------------------------------------------------------------

typedef __attribute__((ext_vector_type(2))) float v2f;
typedef __attribute__((ext_vector_type(8))) float v8f;

__device__ __forceinline__ v8f wmma4(v2f a, v2f b, v8f c) {
  // D(16x16,f32) = A(16x4,f32) * B(4x16,f32) + C
  return __builtin_amdgcn_wmma_f32_16x16x4_f32(false, a, false, b, (short)0, c,
                                               false, false);
}

constexpr int   NPT  = 100000;
constexpr int   KNB  = 16;
constexpr int   CH   = 64;
constexpr int   GR   = 8;
constexpr float EPSV = 1e-5f;

constexpr int NB1 = 3125;   // k1 blocks: 2 row-tiles (of 16) each => 6250 tiles
constexpr int NB2 = 6250;   // k2 blocks: 256 (n,k) pairs each => 1.6M
constexpr int NB4 = 12500;  // k4/k6 blocks: 8 points (1 wave each) => 100000

constexpr size_t NC  = (size_t)NPT * CH;        // 6.4e6 floats
constexpr size_t NKG = (size_t)NPT * KNB * GR;  // 12.8e6 floats

// workspace layout (floats)
constexpr size_t OFF_XQ   = 0;                              // raw feat@wq+bq
constexpr size_t OFF_XK   = OFF_XQ + NC;                    // raw feat@wk+bk
constexpr size_t OFF_VV   = OFF_XK + NC;                    // feat@wv+bv
constexpr size_t OFF_YL   = OFF_VV + NC;                    // raw relation@ww1+bw1 [N,K,8]
constexpr size_t OFF_PQK  = OFF_YL + NKG;                   // k1 partials [NB1][256]
constexpr size_t OFF_PPOS = OFF_PQK + (size_t)NB1 * 256;    // k2 partials [NB2][9]
constexpr size_t OFF_PY   = OFF_PPOS + (size_t)NB2 * 9;     // k4 partials [NB4][16]
constexpr size_t OFF_PRM  = OFF_PY + (size_t)NB4 * 16;      // finalized params (528)
// PRM: 0 qs |64 qsh |128 ks |192 ksh |256 fw0 |320 fw1 |384 fw2 |448 fB |512 wsg[8] |520 wsh[8]
// Total ws: ~33.06M floats = ~132.3 MB (must fit in ws_size).

// ---------------------------------------------------------------------------
// Shared helper: 16x64 = A(16x64) @ wp2(64x64) + bp2, A computed on the fly
// from relative coords with the pos-branch BN folded in. One wave per call.
// Result written to LDS pebOut[16*64] (row-major).
// ---------------------------------------------------------------------------
__device__ __forceinline__ void compute_peb(int lane, float px, float py, float pz,
                                            const float* sFw,    // [256] fw0,fw1,fw2,fB
                                            const float* sWp2T,  // [64*64] wp2 transposed
                                            const float* sBp2,   // [64]
                                            float* pebOut)       // [16*64]
{
  const int m = lane & 15, half = lane >> 4;
  v2f af[16];
#pragma unroll
  for (int t = 0; t < 16; ++t) {
    const int c = 4 * t + 2 * half;
    float a0 = fmaf(px, sFw[c],     fmaf(py, sFw[64 + c],     fmaf(pz, sFw[128 + c],     sFw[192 + c])));
    float a1 = fmaf(px, sFw[c + 1], fmaf(py, sFw[64 + c + 1], fmaf(pz, sFw[128 + c + 1], sFw[192 + c + 1])));
    af[t].x = a0 > 0.f ? a0 : 0.f;
    af[t].y = a1 > 0.f ? a1 : 0.f;
  }
#pragma unroll
  for (int j = 0; j < 4; ++j) {
    const int col = 16 * j + m;
    v8f acc = {0.f, 0.f, 0.f, 0.f, 0.f, 0.f, 0.f, 0.f};
#pragma unroll
    for (int t = 0; t < 16; ++t) {
      v2f b = *(const v2f*)(sWp2T + col * 64 + 4 * t + 2 * half);
      acc = wmma4(af[t], b, acc);
    }
    const float bb = sBp2[col];
#pragma unroll
    for (int r = 0; r < 8; ++r)
      pebOut[(r + 8 * half) * 64 + col] = acc[r] + bb;
  }
}

// ---------------------------------------------------------------------------
// k1: projections via WMMA + deterministic BN partials for Xq/Xk.
// blockDim 64 (2 waves), each wave owns one 16-row tile. 6250 tiles total.
// ---------------------------------------------------------------------------
__device__ __forceinline__ void proj_one(const float* sWT, const v2f* af,
                                         const float* bias, float* outBase,
                                         int row0, int lane,
                                         float* sumSlot, float* sqSlot)
{
  const int m = lane & 15, half = lane >> 4;
#pragma unroll
  for (int j = 0; j < 4; ++j) {
    const int col = 16 * j + m;
    v8f acc = {0.f, 0.f, 0.f, 0.f, 0.f, 0.f, 0.f, 0.f};
#pragma unroll
    for (int t = 0; t < 16; ++t) {
      v2f b = *(const v2f*)(sWT + col * 64 + 4 * t + 2 * half);
      acc = wmma4(af[t], b, acc);
    }
    const float bb = bias[col];
    float ls = 0.f, lq = 0.f;
#pragma unroll
    for (int r = 0; r < 8; ++r) {
      const float v = acc[r] + bb;
      outBase[(size_t)(row0 + r + 8 * half) * 64 + col] = v;
      ls += v;
      lq += v * v;
    }
    if (sumSlot) {
      ls += __shfl_xor(ls, 16, 32);
      lq += __shfl_xor(lq, 16, 32);
      if (half == 0) { sumSlot[col] = ls; sqSlot[col] = lq; }
    }
  }
}

__global__ __launch_bounds__(64) void k1_proj(const float* __restrict__ feat,
                                              const float* __restrict__ wq, const float* __restrict__ bq,
                                              const float* __restrict__ wk, const float* __restrict__ bk,
                                              const float* __restrict__ wv, const float* __restrict__ bv,
                                              float* __restrict__ ws)
{
  __shared__ float sWT[3 * 64 * 64];   // transposed weights: [c][r]
  __shared__ float sSum[2][2][64];     // [wave][mat(q,k)][col]
  __shared__ float sSq[2][2][64];

  const int tid = threadIdx.x;
  for (int i = tid; i < 4096; i += 64) {
    const int r = i >> 6, c = i & 63;
    sWT[c * 64 + r]            = wq[i];
    sWT[4096 + c * 64 + r]     = wk[i];
    sWT[8192 + c * 64 + r]     = wv[i];
  }
  __syncthreads();

  const int w = tid >> 5, lane = tid & 31;
  const int m = lane & 15, half = lane >> 4;
  const int tile = blockIdx.x * 2 + w;     // 3125*2 = 6250 tiles exactly
  const int row0 = tile * 16;

  v2f af[16];
#pragma unroll
  for (int t = 0; t < 16; ++t)
    af[t] = *(const v2f*)(feat + (size_t)(row0 + m) * 64 + 4 * t + 2 * half);

  proj_one(sWT,        af, bq, ws + OFF_XQ, row0, lane, &sSum[w][0][0], &sSq[w][0][0]);
  proj_one(sWT + 4096, af, bk, ws + OFF_XK, row0, lane, &sSum[w][1][0], &sSq[w][1][0]);
  proj_one(sWT + 8192, af, bv, ws + OFF_VV, row0, lane, nullptr, nullptr);

  __syncthreads();
  // 256 partial slots: [sumQ 64 | sqQ 64 | sumK 64 | sqK 64], summed over the
  // 2 waves in fixed order => deterministic.
  for (int s = tid; s < 256; s += 64) {
    const int arr = s >> 6, c = s & 63;
    float v;
    if (arr == 0)      v = sSum[0][0][c] + sSum[1][0][c];
    else if (arr == 1) v = sSq[0][0][c]  + sSq[1][0][c];
    else if (arr == 2) v = sSum[0][1][c] + sSum[1][1][c];
    else               v = sSq[0][1][c]  + sSq[1][1][c];
    ws[OFF_PQK + (size_t)blockIdx.x * 256 + s] = v;
  }
}

// ---------------------------------------------------------------------------
// k2: pos moments: s = sum(pos), M = sum(pos pos^T) over all 1.6M (n,k).
// ---------------------------------------------------------------------------
__global__ __launch_bounds__(256) void k2_posmom(const float* __restrict__ coord,
                                                 const int* __restrict__ ref,
                                                 float* __restrict__ ws)
{
  __shared__ float sP[8][9];
  const int tid = threadIdx.x;
  const int i = blockIdx.x * 256 + tid;      // always < 1.6M (exact)
  const int n = i >> 4;
  const int ri = ref[i];
  const int ic = min(max(ri, 0), NPT - 1);
  const float px = coord[ic * 3 + 0] - coord[n * 3 + 0];
  const float py = coord[ic * 3 + 1] - coord[n * 3 + 1];
  const float pz = coord[ic * 3 + 2] - coord[n * 3 + 2];

  float v[9];
  v[0] = px; v[1] = py; v[2] = pz;
  v[3] = px * px; v[4] = px * py; v[5] = px * pz;
  v[6] = py * py; v[7] = py * pz; v[8] = pz * pz;
#pragma unroll
  for (int d = 1; d < 32; d <<= 1) {
#pragma unroll
    for (int t = 0; t < 9; ++t) v[t] += __shfl_xor(v[t], d, 32);
  }
  const int lane = tid & 31, w = tid >> 5;
  if (lane == 0) {
#pragma unroll
    for (int t = 0; t < 9; ++t) sP[w][t] = v[t];
  }
  __syncthreads();
  if (tid < 9) {
    float s = 0.f;
#pragma unroll 1
    for (int ww = 0; ww < 8; ++ww) s += sP[ww][tid];
    ws[OFF_PPOS + (size_t)blockIdx.x * 9 + tid] = s;
  }
}

// ---------------------------------------------------------------------------
// k3: finalize BN(q), BN(k); closed-form BN for pos-branch, folded into wp1.
// ---------------------------------------------------------------------------
__global__ __launch_bounds__(256) void k3_fin1(const float* __restrict__ gq, const float* __restrict__ betaq,
                                               const float* __restrict__ gk, const float* __restrict__ betak,
                                               const float* __restrict__ wp1, const float* __restrict__ bp1,
                                               const float* __restrict__ gp, const float* __restrict__ betap,
                                               float* __restrict__ ws)
{
  __shared__ float red[256];
  __shared__ float pr[9];
  const int t = threadIdx.x;
  {
    float s = 0.f;
    for (int b = 0; b < NB1; ++b) s += ws[OFF_PQK + (size_t)b * 256 + t];
    red[t] = s;
  }
  if (t < 9) {
    float s = 0.f;
    for (int b = 0; b < NB2; ++b) s += ws[OFF_PPOS + (size_t)b * 9 + t];
    pr[t] = s;
  }
  __syncthreads();
  float* PRM = ws + OFF_PRM;
  if (t < 64) {
    const float invN = 1.f / (float)NPT;
    // q
    {
      const float mm = red[t] * invN;
      const float vv = red[64 + t] * invN - mm * mm;
      const float sc = gq[t] * rsqrtf(vv + EPSV);
      PRM[t] = sc; PRM[64 + t] = betaq[t] - mm * sc;
    }
    // k
    {
      const float mm = red[128 + t] * invN;
      const float vv = red[192 + t] * invN - mm * mm;
      const float sc = gk[t] * rsqrtf(vv + EPSV);
      PRM[128 + t] = sc; PRM[192 + t] = betak[t] - mm * sc;
    }
    // pos branch: x = p.w + b over 1.6M samples, closed form from moments
    {
      const float w0 = wp1[t], w1 = wp1[64 + t], w2 = wp1[128 + t], b = bp1[t];
      const float invNK = 1.f / (float)(NPT * KNB);
      const float sw = pr[0] * w0 + pr[1] * w1 + pr[2] * w2;
      const float mean = sw * invNK + b;
      const float wMw = w0 * w0 * pr[3] + w1 * w1 * pr[6] + w2 * w2 * pr[8]
                      + 2.f * (w0 * w1 * pr[4] + w0 * w2 * pr[5] + w1 * w2 * pr[7]);
      const float ex2 = (wMw + 2.f * b * sw) * invNK + b * b;
      const float var = ex2 - mean * mean;
      const float sc = gp[t] * rsqrtf(var + EPSV);
      const float sh = betap[t] - mean * sc;
      PRM[256 + t] = w0 * sc;          // fw0
      PRM[320 + t] = w1 * sc;          // fw1
      PRM[384 + t] = w2 * sc;          // fw2
      PRM[448 + t] = b * sc + sh;      // fB
    }
  }
}

// ---------------------------------------------------------------------------
// k4: per point (1 wave / point, 8 points / block):
//   q = relu(bn(Xq)); peb via WMMA; y[m][g] = (k[idx]-q+peb) @ ww1 + bw1
//   store y, accumulate per-group sum/sumsq partials.
// ---------------------------------------------------------------------------
__global__ __launch_bounds__(256) void k4_relation(const float* __restrict__ coord,
                                                   const int* __restrict__ ref,
                                                   const float* __restrict__ wp2, const float* __restrict__ bp2,
                                                   const float* __restrict__ ww1, const float* __restrict__ bw1,
                                                   float* __restrict__ ws)
{
  __shared__ float sWp2T[64 * 64];
  __shared__ float sPeb[8][16 * 64];
  __shared__ float sQ[8][64];
  __shared__ float sWw1[64 * 8];
  __shared__ float sFw[256];
  __shared__ float sQS[64], sQSh[64], sKS[64], sKSh[64];
  __shared__ float sBp2[64], sBw1[8];
  __shared__ float sYpart[8][16];

  const float* Xq = ws + OFF_XQ;
  const float* Xk = ws + OFF_XK;
  float* Yl = ws + OFF_YL;
  const float* PRM = ws + OFF_PRM;

  const int tid = threadIdx.x;
  for (int i = tid; i < 4096; i += 256) {
    const int r = i >> 6, c = i & 63;
    sWp2T[c * 64 + r] = wp2[i];
  }
  for (int i = tid; i < 512; i += 256) sWw1[i] = ww1[i];
  sFw[tid] = PRM[256 + tid];
  if (tid < 64) {
    sQS[tid] = PRM[tid];        sQSh[tid] = PRM[64 + tid];
    sKS[tid] = PRM[128 + tid];  sKSh[tid] = PRM[192 + tid];
    sBp2[tid] = bp2[tid];
  }
  if (tid < 8) sBw1[tid] = bw1[tid];
  __syncthreads();

  const int w = tid >> 5, lane = tid & 31;
  const int n = blockIdx.x * 8 + w;   // 12500*8 = 100000 exactly
  const int m = lane & 15, half = lane >> 4;

  // q vector for this point (2 channels per lane)
  {
    int c = lane;
    float x = Xq[(size_t)n * 64 + c];
    sQ[w][c] = fmaxf(fmaf(x, sQS[c], sQSh[c]), 0.f);
    c = lane + 32;
    x = Xq[(size_t)n * 64 + c];
    sQ[w][c] = fmaxf(fmaf(x, sQS[c], sQSh[c]), 0.f);
  }

  const int ri = ref[n * 16 + m];
  const int ic = min(max(ri, 0), NPT - 1);
  const float px = coord[ic * 3 + 0] - coord[n * 3 + 0];
  const float py = coord[ic * 3 + 1] - coord[n * 3 + 1];
  const float pz = coord[ic * 3 + 2] - coord[n * 3 + 2];

  compute_peb(lane, px, py, pz, sFw, sWp2T, sBp2, &sPeb[w][0]);
  __syncthreads();   // peb + q visible

  const int g0 = 4 * half;
  float y0 = sBw1[g0 + 0], y1 = sBw1[g0 + 1], y2 = sBw1[g0 + 2], y3 = sBw1[g0 + 3];
  const float* XkRow = Xk + (size_t)ic * 64;
  const float* pebRow = &sPeb[w][m * 64];
#pragma unroll 4
  for (int c4 = 0; c4 < 64; c4 += 4) {
    const float4 kx = *(const float4*)(XkRow + c4);
    {
      const float kv = fmaxf(fmaf(kx.x, sKS[c4 + 0], sKSh[c4 + 0]), 0.f);
      const float r0 = kv - sQ[w][c4 + 0] + pebRow[c4 + 0];
      y0 = fmaf(r0, sWw1[(c4 + 0) * 8 + g0 + 0], y0);
      y1 = fmaf(r0, sWw1[(c4 + 0) * 8 + g0 + 1], y1);
      y2 = fmaf(r0, sWw1[(c4 + 0) * 8 + g0 + 2], y2);
      y3 = fmaf(r0, sWw1[(c4 + 0) * 8 + g0 + 3], y3);
    }
    {
      const float kv = fmaxf(fmaf(kx.y, sKS[c4 + 1], sKSh[c4 + 1]), 0.f);
      const float r0 = kv - sQ[w][c4 + 1] + pebRow[c4 + 1];
      y0 = fmaf(r0, sWw1[(c4 + 1) * 8 + g0 + 0], y0);
      y1 = fmaf(r0, sWw1[(c4 + 1) * 8 + g0 + 1], y1);
      y2 = fmaf(r0, sWw1[(c4 + 1) * 8 + g0 + 2], y2);
      y3 = fmaf(r0, sWw1[(c4 + 1) * 8 + g0 + 3], y3);
    }
    {
      const float kv = fmaxf(fmaf(kx.z, sKS[c4 + 2], sKSh[c4 + 2]), 0.f);
      const float r0 = kv - sQ[w][c4 + 2] + pebRow[c4 + 2];
      y0 = fmaf(r0, sWw1[(c4 + 2) * 8 + g0 + 0], y0);
      y1 = fmaf(r0, sWw1[(c4 + 2) * 8 + g0 + 1], y1);
      y2 = fmaf(r0, sWw1[(c4 + 2) * 8 + g0 + 2], y2);
      y3 = fmaf(r0, sWw1[(c4 + 2) * 8 + g0 + 3], y3);
    }
    {
      const float kv = fmaxf(fmaf(kx.w, sKS[c4 + 3], sKSh[c4 + 3]), 0.f);
      const float r0 = kv - sQ[w][c4 + 3] + pebRow[c4 + 3];
      y0 = fmaf(r0, sWw1[(c4 + 3) * 8 + g0 + 0], y0);
      y1 = fmaf(r0, sWw1[(c4 + 3) * 8 + g0 + 1], y1);
      y2 = fmaf(r0, sWw1[(c4 + 3) * 8 + g0 + 2], y2);
      y3 = fmaf(r0, sWw1[(c4 + 3) * 8 + g0 + 3], y3);
    }
  }
  // store raw logits (BN(y) applied later in k6)
  {
    float4 yv; yv.x = y0; yv.y = y1; yv.z = y2; yv.w = y3;
    *(float4*)(Yl + ((size_t)n * 16 + m) * 8 + g0) = yv;
  }
  // per-group sum/sumsq over the 16 rows (within each 16-lane half)
  float s0 = y0, s1 = y1, s2 = y2, s3 = y3;
  float q0 = y0 * y0, q1 = y1 * y1, q2 = y2 * y2, q3 = y3 * y3;
#pragma unroll
  for (int d = 1; d < 16; d <<= 1) {
    s0 += __shfl_xor(s0, d, 32); s1 += __shfl_xor(s1, d, 32);
    s2 += __shfl_xor(s2, d, 32); s3 += __shfl_xor(s3, d, 32);
    q0 += __shfl_xor(q0, d, 32); q1 += __shfl_xor(q1, d, 32);
    q2 += __shfl_xor(q2, d, 32); q3 += __shfl_xor(q3, d, 32);
  }
  if (m == 0) {   // lanes 0 (g 0..3) and 16 (g 4..7)
    sYpart[w][g0 + 0] = s0; sYpart[w][g0 + 1] = s1;
    sYpart[w][g0 + 2] = s2; sYpart[w][g0 + 3] = s3;
    sYpart[w][8 + g0 + 0] = q0; sYpart[w][8 + g0 + 1] = q1;
    sYpart[w][8 + g0 + 2] = q2; sYpart[w][8 + g0 + 3] = q3;
  }
  __syncthreads();
  if (tid < 16) {
    float acc = 0.f;
#pragma unroll 1
    for (int ww = 0; ww < 8; ++ww) acc += sYpart[ww][tid];
    ws[OFF_PY + (size_t)blockIdx.x * 16 + tid] = acc;
  }
}

// ---------------------------------------------------------------------------
// k5: finalize BN(y) -> wsg/wsh[8]
// ---------------------------------------------------------------------------
__global__ __launch_bounds__(32) void k5_fin2(const float* __restrict__ gw,
                                              const float* __restrict__ betaw,
                                              float* __restrict__ ws)
{
  __shared__ float yr[16];
  const int t = threadIdx.x;
  if (t < 16) {
    float s = 0.f;
    for (int b = 0; b < NB4; ++b) s += ws[OFF_PY + (size_t)b * 16 + t];
    yr[t] = s;
  }
  __syncthreads();
  if (t < 8) {
    const float invNK = 1.f / (float)(NPT * KNB);
    const float mm = yr[t] * invNK;
    const float ex2 = yr[8 + t] * invNK;
    const float var = ex2 - mm * mm;
    const float sc = gw[t] * rsqrtf(var + EPSV);
    ws[OFF_PRM + 512 + t] = sc;
    ws[OFF_PRM + 520 + t] = betaw[t] - mm * sc;
  }
}

// ---------------------------------------------------------------------------
// k6: per point: peb via WMMA, w = relu(bn(y))@ww2+bw2, softmax over K per
// group (mask AFTER softmax), out[n,c] = sum_k (Vv[idx]+peb)[c] * w[k, c/8].
// ---------------------------------------------------------------------------
__global__ __launch_bounds__(256) void k6_output(const float* __restrict__ coord,
                                                 const int* __restrict__ ref,
                                                 const float* __restrict__ wp2, const float* __restrict__ bp2,
                                                 const float* __restrict__ ww2, const float* __restrict__ bw2,
                                                 const float* __restrict__ ws, float* __restrict__ out)
{
  __shared__ float sWp2T[64 * 64];
  __shared__ float sPeb[8][16 * 64];
  __shared__ float sW[8][16 * 8];
  __shared__ int   sIdx[8][16];
  __shared__ float sFw[256], sBp2[64], sWw2[64], sBw2[8], sWSg[8], sWSh[8];

  const float* Vv = ws + OFF_VV;
  const float* Yl = ws + OFF_YL;
  const float* PRM = ws + OFF_PRM;

  const int tid = threadIdx.x;
  for (int i = tid; i < 4096; i += 256) {
    const int r = i >> 6, c = i & 63;
    sWp2T[c * 64 + r] = wp2[i];
  }
  sFw[tid] = PRM[256 + tid];
  if (tid < 64) { sBp2[tid] = bp2[tid]; sWw2[tid] = ww2[tid]; }
  if (tid < 8) { sBw2[tid] = bw2[tid]; sWSg[tid] = PRM[512 + tid]; sWSh[tid] = PRM[520 + tid]; }
  __syncthreads();

  const int w = tid >> 5, lane = tid & 31;
  const int n = blockIdx.x * 8 + w;
  const int m = lane & 15, half = lane >> 4;

  const int ri = ref[n * 16 + m];
  const int ic = min(max(ri, 0), NPT - 1);
  const float mask = (ri >= 0) ? 1.f : 0.f;
  if (half == 0) sIdx[w][m] = ic;

  const float px = coord[ic * 3 + 0] - coord[n * 3 + 0];
  const float py = coord[ic * 3 + 1] - coord[n * 3 + 1];
  const float pz = coord[ic * 3 + 2] - coord[n * 3 + 2];

  compute_peb(lane, px, py, pz, sFw, sWp2T, sBp2, &sPeb[w][0]);

  // w logits: z = relu(bn(y)) (each lane holds 4 groups), wl = z@ww2 across halves
  const int g0 = 4 * half;
  const float4 ya = *(const float4*)(Yl + ((size_t)n * 16 + m) * 8 + g0);
  const float z0 = fmaxf(fmaf(ya.x, sWSg[g0 + 0], sWSh[g0 + 0]), 0.f);
  const float z1 = fmaxf(fmaf(ya.y, sWSg[g0 + 1], sWSh[g0 + 1]), 0.f);
  const float z2 = fmaxf(fmaf(ya.z, sWSg[g0 + 2], sWSh[g0 + 2]), 0.f);
  const float z3 = fmaxf(fmaf(ya.w, sWSg[g0 + 3], sWSh[g0 + 3]), 0.f);
  float wl[8];
#pragma unroll
  for (int gp = 0; gp < 8; ++gp) {
    float v = z0 * sWw2[(g0 + 0) * 8 + gp];
    v = fmaf(z1, sWw2[(g0 + 1) * 8 + gp], v);
    v = fmaf(z2, sWw2[(g0 + 2) * 8 + gp], v);
    v = fmaf(z3, sWw2[(g0 + 3) * 8 + gp], v);
    wl[gp] = v;
  }
#pragma unroll
  for (int gp = 0; gp < 8; ++gp) {
    wl[gp] += __shfl_xor(wl[gp], 16, 32);   // combine the two half-sums
    wl[gp] += sBw2[gp];
  }
  // softmax over the 16 neighbors (lanes within each half), then mask
#pragma unroll
  for (int gp = 0; gp < 8; ++gp) {
    float mx = wl[gp];
#pragma unroll
    for (int d = 1; d < 16; d <<= 1) mx = fmaxf(mx, __shfl_xor(mx, d, 32));
    const float e = expf(wl[gp] - mx);
    float sm = e;
#pragma unroll
    for (int d = 1; d < 16; d <<= 1) sm += __shfl_xor(sm, d, 32);
    wl[gp] = e / sm * mask;
  }
  if (half == 0) {
#pragma unroll
    for (int gp = 0; gp < 8; ++gp) sW[w][m * 8 + gp] = wl[gp];
  }
  __syncthreads();   // peb, sW, sIdx visible

  // out[n][c] = sum_m (Vv[idx_m][c] + peb[m][c]) * w[m][c/8]; 2 channels/lane
  const int c0 = lane, c1 = lane + 32;
  float o0 = 0.f, o1 = 0.f;
#pragma unroll 1
  for (int mm = 0; mm < 16; ++mm) {
    const int ici = sIdx[w][mm];
    const float* vrow = Vv + (size_t)ici * 64;
    const float w0 = sW[w][mm * 8 + (c0 >> 3)];
    const float w1 = sW[w][mm * 8 + (c1 >> 3)];
    o0 = fmaf(vrow[c0] + sPeb[w][mm * 64 + c0], w0, o0);
    o1 = fmaf(vrow[c1] + sPeb[w][mm * 64 + c1], w1, o1);
  }
  out[(size_t)n * 64 + c0] = o0;
  out[(size_t)n * 64 + c1] = o1;
}

// ---------------------------------------------------------------------------
extern "C" void kernel_launch(void* const* d_in, const int* in_sizes, int n_in,
                              void* d_out, int out_size, void* d_ws, size_t ws_size,
                              hipStream_t stream)
{
  const float* feat  = (const float*)d_in[0];
  const float* coord = (const float*)d_in[1];
  const int*   ref   = (const int*)d_in[2];
  const float* wq = (const float*)d_in[3],  *bq = (const float*)d_in[4];
  const float* gq = (const float*)d_in[5],  *betaq = (const float*)d_in[6];
  const float* wk = (const float*)d_in[7],  *bk = (const float*)d_in[8];
  const float* gk = (const float*)d_in[9],  *betak = (const float*)d_in[10];
  const float* wv = (const float*)d_in[11], *bv = (const float*)d_in[12];
  const float* wp1 = (const float*)d_in[13], *bp1 = (const float*)d_in[14];
  const float* gp = (const float*)d_in[15],  *betap = (const float*)d_in[16];
  const float* wp2 = (const float*)d_in[17], *bp2 = (const float*)d_in[18];
  const float* ww1 = (const float*)d_in[19], *bw1 = (const float*)d_in[20];
  const float* gw = (const float*)d_in[21],  *betaw = (const float*)d_in[22];
  const float* ww2 = (const float*)d_in[23], *bw2 = (const float*)d_in[24];

  float* ws = (float*)d_ws;
  float* outp = (float*)d_out;

  (void)in_sizes; (void)n_in; (void)out_size; (void)ws_size;

  k1_proj<<<NB1, 64, 0, stream>>>(feat, wq, bq, wk, bk, wv, bv, ws);
  k2_posmom<<<NB2, 256, 0, stream>>>(coord, ref, ws);
  k3_fin1<<<1, 256, 0, stream>>>(gq, betaq, gk, betak, wp1, bp1, gp, betap, ws);
  k4_relation<<<NB4, 256, 0, stream>>>(coord, ref, wp2, bp2, ww1, bw1, ws);
  k5_fin2<<<1, 32, 0, stream>>>(gw, betaw, ws);
  k6_output<<<NB4, 256, 0, stream>>>(coord, ref, wp2, bp2, ww2, bw2, ws, outp);
}